// EnhancedOFTQKVLayer_46411416601018
// MI455X (gfx1250) — compile-verified
//
#include <hip/hip_runtime.h>

#define HIDDEN   1024
#define OUTDIM   3072
#define MROWS    32768        // 8 * 4096
#define NBLK     8
#define BLKSZ    128
#define EPSV     1e-6f

typedef __attribute__((ext_vector_type(2))) float v2f;
typedef __attribute__((ext_vector_type(4))) float f32x4;
typedef __attribute__((ext_vector_type(8))) float v8f;
typedef __attribute__((ext_vector_type(4))) int   v4i;

// async global->LDS builtin takes (v4i AS1*, v4i AS3*, imm offset, imm cpol)
typedef v4i __attribute__((address_space(1)))* gptr_v4i;
typedef v4i __attribute__((address_space(3)))* lptr_v4i;

#if defined(__has_builtin)
#if __has_builtin(__builtin_amdgcn_global_load_async_to_lds_b128)
#define HAVE_ASYNC_LDS 1
#endif
#endif

__device__ __forceinline__ void wait_asynccnt0() {
#if defined(__has_builtin)
#if __has_builtin(__builtin_amdgcn_s_wait_asynccnt)
  __builtin_amdgcn_s_wait_asynccnt(0);
  return;
#else
  asm volatile("s_wait_asynccnt 0x0" ::: "memory");
  return;
#endif
#else
  asm volatile("s_wait_asynccnt 0x0" ::: "memory");
#endif
}

// ---------------------------------------------------------------------------
// Kernel 1: Cayley transform. One workgroup per (qkv, block) pair (24 total).
// G = I + S + eps*I inverted in LDS via Gauss-Jordan (no pivoting needed:
// I + S with skew-symmetric S has symmetric part I, always well conditioned).
// Then orth = (I - S) @ G^{-1}.
// ---------------------------------------------------------------------------
__global__ __launch_bounds__(256) void cayley_kernel(
    const float* __restrict__ qR, const float* __restrict__ kR,
    const float* __restrict__ vR, float* __restrict__ orth) {
  __shared__ float G[BLKSZ][BLKSZ + 2];
  __shared__ float V[BLKSZ][BLKSZ + 2];
  __shared__ float pivinv;

  const int g   = blockIdx.x;          // 0..23
  const int qkv = g >> 3;
  const int n   = g & 7;
  const float* A = (qkv == 0 ? qR : (qkv == 1 ? kR : vR)) + (size_t)n * BLKSZ * BLKSZ;
  const int tid = threadIdx.x;

  for (int idx = tid; idx < BLKSZ * BLKSZ; idx += 256) {
    const int i = idx >> 7, j = idx & 127;
    const float s = 0.5f * (A[i * BLKSZ + j] - A[j * BLKSZ + i]);
    G[i][j] = s + ((i == j) ? (1.0f + EPSV) : 0.0f);
    V[i][j] = (i == j) ? 1.0f : 0.0f;
  }
  __syncthreads();

  const int r    = tid & 127;
  const int half = tid >> 7;
  for (int p = 0; p < BLKSZ; ++p) {
    if (tid == 0) pivinv = 1.0f / G[p][p];
    __syncthreads();
    if (half == 0) G[p][r] *= pivinv; else V[p][r] *= pivinv;
    __syncthreads();
    const float f = G[r][p];           // read factor before anyone clobbers it
    __syncthreads();
    if (r != p) {
      if (half == 0) {
        for (int c = 0; c < BLKSZ; ++c) G[r][c] -= f * G[p][c];
      } else {
        for (int c = 0; c < BLKSZ; ++c) V[r][c] -= f * V[p][c];
      }
    }
    __syncthreads();
  }

  // orth = (I - S) @ V
  float* O = orth + (size_t)g * BLKSZ * BLKSZ;
  for (int idx = tid; idx < BLKSZ * BLKSZ; idx += 256) {
    const int i = idx >> 7, j = idx & 127;
    float sum = V[i][j];
    for (int k = 0; k < BLKSZ; ++k) {
      const float s_ik = 0.5f * (A[i * BLKSZ + k] - A[k * BLKSZ + i]);
      sum -= s_ik * V[k][j];
    }
    O[i * BLKSZ + j] = sum;
  }
}

// ---------------------------------------------------------------------------
// Kernel 2: fold orth blocks into weights.
// filt[o, n*128 + c] = sum_b W[o, n*128 + b] * orth[qkv*8+n][b][c]
// ---------------------------------------------------------------------------
__global__ __launch_bounds__(256) void fold_kernel(
    const float* __restrict__ W,      // [3072][1024]
    const float* __restrict__ orth,   // [24][128][128]
    float* __restrict__ filt) {       // [3072][1024]
  __shared__ float Rl[BLKSZ][BLKSZ];
  __shared__ float Wl[32][BLKSZ];

  const int o_base = blockIdx.x * 32;
  const int n      = blockIdx.y;
  const int qkv    = o_base >> 10;
  const float* R   = orth + (size_t)(qkv * NBLK + n) * BLKSZ * BLKSZ;
  const int tid    = threadIdx.x;

  for (int idx = tid; idx < BLKSZ * BLKSZ; idx += 256) ((float*)Rl)[idx] = R[idx];
  for (int idx = tid; idx < 32 * BLKSZ; idx += 256) {
    const int ro = idx >> 7, b = idx & 127;
    Wl[ro][b] = W[(size_t)(o_base + ro) * HIDDEN + n * BLKSZ + b];
  }
  __syncthreads();

  for (int idx = tid; idx < 32 * BLKSZ; idx += 256) {
    const int ro = idx >> 7, c = idx & 127;
    float sum = 0.0f;
    for (int b = 0; b < BLKSZ; ++b) sum += Wl[ro][b] * Rl[b][c];
    filt[(size_t)(o_base + ro) * HIDDEN + n * BLKSZ + c] = sum;
  }
}

// ---------------------------------------------------------------------------
// Kernel 3: main GEMM  out[M,N] = x[M,K] @ filt[N,K]^T + bias[N]
// Workgroup tile 128(M) x 256(N), K-chunks of 32, 8 waves in a 2x4 grid,
// each wave owns a 64x64 tile = 4x4 accumulators of v_wmma_f32_16x16x4_f32.
// Double-buffered LDS fed by GLOBAL_LOAD_ASYNC_TO_LDS_B128 (ASYNCcnt) where
// the toolchain exposes it; synchronous b128 staging otherwise.
// LDS row stride 36 floats: distinct banks for the 16 fragment rows and
// 16B alignment for b128 transfers / 8B alignment for v2f fragment loads.
// ---------------------------------------------------------------------------
#define TM  128
#define TN  256
#define TK  32
#define LDT 36

__global__ __launch_bounds__(256) void gemm_kernel(
    const float* __restrict__ x,      // [32768][1024]
    const float* __restrict__ filt,   // [3072][1024]
    const float* __restrict__ bias,   // [3072]
    float* __restrict__ out) {        // [32768][3072]
  __shared__ float As[2][TM][LDT];
  __shared__ float Bs[2][TN][LDT];

  const int m_base = blockIdx.y * TM;
  const int n_base = blockIdx.x * TN;
  const int tid   = threadIdx.x;
  const int wave  = tid >> 5;
  const int lane  = tid & 31;
  const int wm    = wave & 1;      // 0..1  -> 64-row slab
  const int wn    = wave >> 1;     // 0..3  -> 64-col slab
  const int lidx  = lane & 15;
  const int lhalf = lane >> 4;

  v8f acc[4][4];
#pragma unroll
  for (int i = 0; i < 4; ++i)
#pragma unroll
    for (int j = 0; j < 4; ++j) acc[i][j] = v8f{};

  // staging assignment: A -> thread covers 16 floats of one row,
  //                     B -> thread covers one full 32-float row
  const int a_row = tid >> 1;              // 0..127
  const int a_col = (tid & 1) * 16;        // 0 / 16
  const float* xrow = x    + (size_t)(m_base + a_row) * HIDDEN + a_col;
  const float* frow = filt + (size_t)(n_base + tid)   * HIDDEN;

#if HAVE_ASYNC_LDS
#define ISSUE_CHUNK(buf, k0)                                                   \
  do {                                                                         \
    _Pragma("unroll")                                                          \
    for (int i_ = 0; i_ < 4; ++i_)                                             \
      __builtin_amdgcn_global_load_async_to_lds_b128(                          \
          (gptr_v4i)(xrow + (k0) + i_ * 4),                                    \
          (lptr_v4i)&As[buf][a_row][a_col + i_ * 4], 0, 0);                    \
    _Pragma("unroll")                                                          \
    for (int i_ = 0; i_ < 8; ++i_)                                             \
      __builtin_amdgcn_global_load_async_to_lds_b128(                          \
          (gptr_v4i)(frow + (k0) + i_ * 4),                                    \
          (lptr_v4i)&Bs[buf][tid][i_ * 4], 0, 0);                              \
  } while (0)
#else
#define ISSUE_CHUNK(buf, k0)                                                   \
  do {                                                                         \
    _Pragma("unroll")                                                          \
    for (int i_ = 0; i_ < 4; ++i_)                                             \
      *(f32x4*)&As[buf][a_row][a_col + i_ * 4] =                               \
          *(const f32x4*)(xrow + (k0) + i_ * 4);                               \
    _Pragma("unroll")                                                          \
    for (int i_ = 0; i_ < 8; ++i_)                                             \
      *(f32x4*)&Bs[buf][tid][i_ * 4] = *(const f32x4*)(frow + (k0) + i_ * 4);  \
  } while (0)
#endif

  // Prologue: fetch chunk 0 into buffer 0
  ISSUE_CHUNK(0, 0);
#if HAVE_ASYNC_LDS
  wait_asynccnt0();
#endif
  __syncthreads();

  for (int k0 = 0; k0 < HIDDEN; k0 += TK) {
    const int cur = (k0 / TK) & 1;
    const int nxt = cur ^ 1;
    if (k0 + TK < HIDDEN) ISSUE_CHUNK(nxt, k0 + TK);

    for (int kk = 0; kk < TK; kk += 4) {
      v2f a[4], b[4];
#pragma unroll
      for (int i = 0; i < 4; ++i)
        a[i] = *(const v2f*)&As[cur][wm * 64 + i * 16 + lidx][kk + lhalf * 2];
#pragma unroll
      for (int j = 0; j < 4; ++j)
        b[j] = *(const v2f*)&Bs[cur][wn * 64 + j * 16 + lidx][kk + lhalf * 2];
#pragma unroll
      for (int i = 0; i < 4; ++i)
#pragma unroll
        for (int j = 0; j < 4; ++j)
          acc[i][j] = __builtin_amdgcn_wmma_f32_16x16x4_f32(
              false, a[i], false, b[j], (short)0, acc[i][j], false, false);
    }

#if HAVE_ASYNC_LDS
    wait_asynccnt0();    // next-chunk tiles have landed in LDS
#endif
    __syncthreads();     // everyone done reading cur / writing nxt
  }

  // Epilogue: C layout — VGPR r: M = r + 8*lhalf, N = lidx
#pragma unroll
  for (int j = 0; j < 4; ++j) {
    const int N0 = n_base + wn * 64 + j * 16;
    const float bv = bias[N0 + lidx];
#pragma unroll
    for (int i = 0; i < 4; ++i) {
      const int M0 = m_base + wm * 64 + i * 16;
      float* op = out + (size_t)(M0 + lhalf * 8) * OUTDIM + N0 + lidx;
#pragma unroll
      for (int rr = 0; rr < 8; ++rr)
        op[(size_t)rr * OUTDIM] = acc[i][j][rr] + bv;
    }
  }
#undef ISSUE_CHUNK
}

// ---------------------------------------------------------------------------
extern "C" void kernel_launch(void* const* d_in, const int* in_sizes, int n_in,
                              void* d_out, int out_size, void* d_ws, size_t ws_size,
                              hipStream_t stream) {
  const float* attn_weight = (const float*)d_in[0];  // [3072*1024]
  const float* bias        = (const float*)d_in[1];  // [3072]
  const float* x           = (const float*)d_in[2];  // [8*4096*1024]
  const float* q_R         = (const float*)d_in[3];  // [8*128*128]
  const float* k_R         = (const float*)d_in[4];
  const float* v_R         = (const float*)d_in[5];
  float* out = (float*)d_out;                        // [32768*3072]

  float* filt = (float*)d_ws;                        // 3072*1024 floats
  float* orth = filt + (size_t)OUTDIM * HIDDEN;      // 24*128*128 floats

  cayley_kernel<<<dim3(24), dim3(256), 0, stream>>>(q_R, k_R, v_R, orth);
  fold_kernel<<<dim3(96, 8), dim3(256), 0, stream>>>(attn_weight, orth, filt);
  gemm_kernel<<<dim3(OUTDIM / TN, MROWS / TM), dim3(256), 0, stream>>>(x, filt, bias, out);
}